// EncoderRNN_37881611551124
// MI455X (gfx1250) — compile-verified
//
#include <hip/hip_runtime.h>
#include <hip/hip_bf16.h>
#include <math.h>

// GRU encoder for MI455X (gfx1250, wave32, WMMA + async global->LDS).
// B=32, T=512, H=1024, V=32000. All math fp32 via V_WMMA_F32_16X16X4_F32.

#define BB 32
#define TT 512
#define HH 1024
#define GG (3 * HH)      // 3072
#define BT (BB * TT)     // 16384

#define KCH 32           // K-chunk per pipeline stage
#define KSEG (KCH / 4)   // 16-byte segments per row per chunk (8)
#define LAP (KCH + 4)    // padded LDS row pitch (36 floats = 144B, 16B aligned)

typedef __attribute__((ext_vector_type(2))) float v2f;
typedef __attribute__((ext_vector_type(8))) float v8f;
typedef __attribute__((ext_vector_type(4))) int   v4i;

#define GLOBAL_AS __attribute__((address_space(1)))
#define LDS_AS    __attribute__((address_space(3)))

#if defined(__AMDGCN__) && __has_builtin(__builtin_amdgcn_global_load_async_to_lds_b128)
#define HAS_ASYNC_LDS 1
__device__ __forceinline__ void async_copy16(const float* g, float* l) {
    // 16 bytes per lane, global -> LDS via the async engine (ASYNCcnt).
    // Builtin signature (from clang diagnostic): (AS1 v4i*, AS3 v4i*, imm, imm)
    __builtin_amdgcn_global_load_async_to_lds_b128(
        (GLOBAL_AS v4i*)(g), (LDS_AS v4i*)(l), 0, 0);
}
__device__ __forceinline__ void async_wait0() {
#if __has_builtin(__builtin_amdgcn_s_wait_asynccnt)
    __builtin_amdgcn_s_wait_asynccnt(0);
#else
    asm volatile("s_wait_asynccnt 0x0" ::: "memory");
#endif
}
#else
#define HAS_ASYNC_LDS 0
#endif

// ---------------------------------------------------------------------------
// Kernel 0: zero the initial hidden state (workspace is poisoned 0xAA).
// ---------------------------------------------------------------------------
__global__ void gru_zero_h0(float* __restrict__ h0) {
    int i = blockIdx.x * blockDim.x + threadIdx.x;
    if (i < BB * HH) h0[i] = 0.0f;
}

// ---------------------------------------------------------------------------
// Kernel 1: xg[(b*T+t), g] = sum_h emb[x[b,t], h] * w_ih[g, h] + b_ih[g]
// Block: 256 threads (8 waves). Tile: 16 rows x 128 cols. Double-buffered
// async K-chunks of 32. Each wave computes one 16x16 f32 WMMA tile.
// ---------------------------------------------------------------------------
__global__ void __launch_bounds__(256)
gru_xg_gemm(const int* __restrict__ x, const float* __restrict__ emb,
            const float* __restrict__ w_ih, const float* __restrict__ b_ih,
            float* __restrict__ xg) {
    __shared__ int   tok[16];
    __shared__ float lA[2][16][LAP];     // gathered emb rows, [m][k]
    __shared__ float lB[2][128][LAP];    // w_ih rows (output cols), [n][k]

    const int tid  = threadIdx.x;
    const int wv   = tid >> 5;           // wave 0..7 -> N subtile
    const int lane = tid & 31;
    const int lm   = lane & 15;          // M (A) / N (B) index in fragment
    const int lh   = lane >> 4;          // K half-select (0/1)

    const int m0 = blockIdx.x * 16;      // flat (b*T+t) row tile
    const int n0 = blockIdx.y * 128;     // gate-column tile

    if (tid < 16) tok[tid] = x[m0 + tid];
    __syncthreads();

    const int NCH = HH / KCH;            // 32 chunks

    auto stage = [&](int buf, int k0) {
#if HAS_ASYNC_LDS
        if (tid < 16 * KSEG) {           // 128 segs: one per low thread
            int m = tid >> 3, s = tid & 7;
            async_copy16(&emb[(size_t)tok[m] * HH + k0 + s * 4],
                         &lA[buf][m][s * 4]);
        }
        for (int i = tid; i < 128 * KSEG; i += 256) {   // 1024 segs
            int n = i >> 3, s = i & 7;
            async_copy16(&w_ih[(size_t)(n0 + n) * HH + k0 + s * 4],
                         &lB[buf][n][s * 4]);
        }
#else
        for (int i = tid; i < 16 * KCH; i += 256) {
            int m = i / KCH, k = i % KCH;
            lA[buf][m][k] = emb[(size_t)tok[m] * HH + k0 + k];
        }
        for (int i = tid; i < 128 * KCH; i += 256) {
            int n = i / KCH, k = i % KCH;
            lB[buf][n][k] = w_ih[(size_t)(n0 + n) * HH + k0 + k];
        }
#endif
    };
    auto settle = [&]() {
#if HAS_ASYNC_LDS
        async_wait0();
#endif
        __syncthreads();
    };

    v8f acc = {};
    stage(0, 0);
    settle();
    for (int kc = 0; kc < NCH; ++kc) {
        const int buf = kc & 1;
        if (kc + 1 < NCH) stage(buf ^ 1, (kc + 1) * KCH);   // overlap copy
        for (int kk = 0; kk < KCH; kk += 4) {
            v2f a, b;
            a.x = lA[buf][lm][kk + 2 * lh + 0];
            a.y = lA[buf][lm][kk + 2 * lh + 1];
            b.x = lB[buf][wv * 16 + lm][kk + 2 * lh + 0];
            b.y = lB[buf][wv * 16 + lm][kk + 2 * lh + 1];
            acc = __builtin_amdgcn_wmma_f32_16x16x4_f32(
                false, a, false, b, (short)0, acc, false, false);
        }
        if (kc + 1 < NCH) settle();
    }

    const int col  = n0 + wv * 16 + lm;
    const float bi = b_ih[col];
#pragma unroll
    for (int r = 0; r < 8; ++r) {
        int row = m0 + r + lh * 8;       // C layout: vgpr r -> M=r / M=r+8
        xg[(size_t)row * GG + col] = acc[r] + bi;
    }
}

// ---------------------------------------------------------------------------
// Kernel 2: one recurrent step. Grid: 32 blocks, each owns 32 hidden columns
// of ALL three gates (so gating fuses in-block). Block: 384 threads = 12
// waves = 2 M-tiles x 2 N-tiles x 3 gates, one 16x16 WMMA tile per wave.
// Async double-buffered K-chunks of 32.
// ---------------------------------------------------------------------------
__global__ void __launch_bounds__(384)
gru_step(int t, const float* __restrict__ hprev, float* __restrict__ hnew,
         const float* __restrict__ xg, const float* __restrict__ w_hh,
         const float* __restrict__ b_hh, float* __restrict__ all_hidden,
         float* __restrict__ last_hidden) {
    __shared__ float lA[2][32][LAP];     // h_prev rows, [m][k]
    __shared__ float lB[2][96][LAP];     // w_hh rows (3x32 gate cols), [n][k]
    __shared__ float lC[32][100];        // hg staging: 32 rows x 96 cols

    const int tid  = threadIdx.x;
    const int wv   = tid >> 5;           // 0..11
    const int lane = tid & 31;
    const int lm   = lane & 15;
    const int lh   = lane >> 4;
    const int mt   = wv & 1;             // M tile (batch 0-15 / 16-31)
    const int nt   = (wv >> 1) & 1;      // N subtile within the 32-col slice
    const int gate = wv >> 2;            // 0=r, 1=z, 2=n

    const int col0 = blockIdx.x * 32;    // hidden-column slice of this block
    const int NCH  = HH / KCH;           // 32 chunks

    auto stage = [&](int buf, int k0) {
#if HAS_ASYNC_LDS
        for (int i = tid; i < 32 * KSEG; i += 384) {    // 256 segs
            int m = i >> 3, s = i & 7;
            async_copy16(&hprev[m * HH + k0 + s * 4], &lA[buf][m][s * 4]);
        }
        for (int i = tid; i < 96 * KSEG; i += 384) {    // 768 segs
            int n = i >> 3, s = i & 7;
            int grow = (n / 32) * HH + col0 + (n % 32); // row of w_hh
            async_copy16(&w_hh[(size_t)grow * HH + k0 + s * 4],
                         &lB[buf][n][s * 4]);
        }
#else
        for (int i = tid; i < 32 * KCH; i += 384) {
            int m = i / KCH, k = i % KCH;
            lA[buf][m][k] = hprev[m * HH + k0 + k];
        }
        for (int i = tid; i < 96 * KCH; i += 384) {
            int n = i / KCH, k = i % KCH;
            int grow = (n / 32) * HH + col0 + (n % 32);
            lB[buf][n][k] = w_hh[(size_t)grow * HH + k0 + k];
        }
#endif
    };
    auto settle = [&]() {
#if HAS_ASYNC_LDS
        async_wait0();
#endif
        __syncthreads();
    };

    const int nIdx = gate * 32 + nt * 16 + lm;          // local B row
    v8f acc = {};
    stage(0, 0);
    settle();
    for (int kc = 0; kc < NCH; ++kc) {
        const int buf = kc & 1;
        if (kc + 1 < NCH) stage(buf ^ 1, (kc + 1) * KCH);
        for (int kk = 0; kk < KCH; kk += 4) {
            v2f a, b;
            a.x = lA[buf][mt * 16 + lm][kk + 2 * lh + 0];
            a.y = lA[buf][mt * 16 + lm][kk + 2 * lh + 1];
            b.x = lB[buf][nIdx][kk + 2 * lh + 0];
            b.y = lB[buf][nIdx][kk + 2 * lh + 1];
            acc = __builtin_amdgcn_wmma_f32_16x16x4_f32(
                false, a, false, b, (short)0, acc, false, false);
        }
        if (kc + 1 < NCH) settle();
    }

    // Dump hg (+ b_hh) into LDS for the fused gating stage.
    {
        const float bh = b_hh[gate * HH + col0 + nt * 16 + lm];
#pragma unroll
        for (int r = 0; r < 8; ++r) {
            lC[mt * 16 + r + lh * 8][nIdx] = acc[r] + bh;
        }
    }
    __syncthreads();

    // Fused gating on the 32x32 (batch x hidden-col) slice.
    for (int i = tid; i < 32 * 32; i += 384) {
        int b = i >> 5, c = i & 31;
        int col = col0 + c;
        float hr = lC[b][c], hz = lC[b][32 + c], hn = lC[b][64 + c];
        size_t xrow = ((size_t)b * TT + t) * GG;
        float xr = xg[xrow + col];
        float xz = xg[xrow + HH + col];
        float xn = xg[xrow + 2 * HH + col];
        float rg = 1.0f / (1.0f + __expf(-(xr + hr)));
        float zg = 1.0f / (1.0f + __expf(-(xz + hz)));
        float ng = tanhf(xn + rg * hn);
        float hp = hprev[b * HH + col];
        float hv = (1.0f - zg) * ng + zg * hp;
        hnew[b * HH + col] = hv;
        all_hidden[((size_t)b * TT + t) * HH + col] = hv;
        if (t == TT - 1) last_hidden[b * HH + col] = hv;
    }
}

// ---------------------------------------------------------------------------
// Host launcher. Inputs: x(i32), emb, w_ih, w_hh, b_ih, b_hh (f32).
// Output: all_hidden (B*T*H) ++ last_hidden (B*H), f32.
// Workspace: xg (B*T*3H f32, ~201 MB) + 2x h ping-pong (B*H f32 each).
// ---------------------------------------------------------------------------
extern "C" void kernel_launch(void* const* d_in, const int* in_sizes, int n_in,
                              void* d_out, int out_size, void* d_ws, size_t ws_size,
                              hipStream_t stream) {
    const int*   x    = (const int*)d_in[0];
    const float* emb  = (const float*)d_in[1];
    const float* w_ih = (const float*)d_in[2];
    const float* w_hh = (const float*)d_in[3];
    const float* b_ih = (const float*)d_in[4];
    const float* b_hh = (const float*)d_in[5];

    float* all_hidden  = (float*)d_out;                    // (B, T, H)
    float* last_hidden = all_hidden + (size_t)BT * HH;     // (1, B, H)

    float* xg   = (float*)d_ws;                            // (B*T, 3H)
    float* hbuf = xg + (size_t)BT * GG;                    // 2 x (B, H)

    // h0 = 0 (workspace is poisoned, and we must be replay-deterministic).
    gru_zero_h0<<<(BB * HH + 255) / 256, 256, 0, stream>>>(hbuf);

    // Input-gate GEMM: 1024 row-tiles x 24 col-tiles.
    gru_xg_gemm<<<dim3(BT / 16, GG / 128), 256, 0, stream>>>(x, emb, w_ih, b_ih, xg);

    // Sequential recurrence: 512 graph nodes, h ping-pongs in workspace.
    for (int t = 0; t < TT; ++t) {
        float* hp = hbuf + (size_t)(t & 1) * BB * HH;
        float* hn = hbuf + (size_t)((t + 1) & 1) * BB * HH;
        gru_step<<<HH / 32, 384, 0, stream>>>(t, hp, hn, xg, w_hh, b_hh,
                                              all_hidden, last_hidden);
    }
}